// LRAttnDecoderRNN_40767829574374
// MI455X (gfx1250) — compile-verified
//
#include <hip/hip_runtime.h>
#include <hip/hip_bf16.h>
#include <math.h>

// ---------------------------------------------------------------------------
// Model dims (fixed by the reference)
// ---------------------------------------------------------------------------
#define NB      64      // batch
#define T_ENC   2048
#define H_ENC   512
#define H_ATT   256
#define H_DEC   512
#define O_DEC   80
#define SPKR    64
#define C_DCA   16
#define DCA_K   21
#define PRIOR_K 11
#define PREV_K  31
#define LSTM_H  2048    // 4*H_DEC
#define D_T     512
#define FF_DIM  1024

#define KT      32      // k panel staged per TDM transfer
#define ARS     34      // padded LDS row stride (dwords) for the A panel

typedef __attribute__((ext_vector_type(2))) float v2f;
typedef __attribute__((ext_vector_type(8))) float v8f;
typedef __attribute__((ext_vector_type(4))) unsigned int v4u;
typedef __attribute__((ext_vector_type(8))) int v8i_t;
typedef __attribute__((ext_vector_type(4))) int v4i_t;

enum { ACT_NONE = 0, ACT_RELU = 1, ACT_TANH = 2 };

__device__ __forceinline__ float sigmf(float x) { return 1.0f / (1.0f + expf(-x)); }

__device__ __forceinline__ unsigned lds_off(const void* p)
{
    // LDS aperture: addr[63:32] selects the aperture, addr[31:0] is the LDS offset.
    return (unsigned)(unsigned long long)(uintptr_t)p;
}

// ---------------------------------------------------------------------------
// Build a 2-D TDM descriptor (D#) and issue TENSOR_LOAD_TO_LDS.
// D# group0: {count=1, lds_addr, global_addr[56:0], type=2}
// D# group1: {wg_mask=0, data_size=4B, pad ctl, tensor_dim0/1, tile_dim0/1,
//             tensor_dim0_stride}; groups 2/3 zero (<=2D tensor).
// Tile: tile_d1 rows of tile_d0 contiguous fp32 elements, row stride = stride
// elements. Optional LDS padding: pad_amount dwords after every pad_interval
// dwords stored (codes per ISA 8.4).
// ---------------------------------------------------------------------------
__device__ __forceinline__ void tdm_load_2d(unsigned lds_addr, const float* gptr,
                                            unsigned tile_d0, unsigned tile_d1,
                                            unsigned long long stride,
                                            int pad_en, int pad_interval_code,
                                            int pad_amount_code)
{
    unsigned long long ga = (unsigned long long)(uintptr_t)gptr;
    v4u g0;
    g0.x = 1u;                                                   // count = 1
    g0.y = lds_addr;                                             // LDS byte addr
    g0.z = (unsigned)(ga & 0xffffffffu);                         // global addr lo
    g0.w = (unsigned)((ga >> 32) & 0x01ffffffu) | (2u << 30);    // addr hi | type=2

    const unsigned td0 = tile_d0;   // tensor dims sized to the tile: no OOB clipping
    const unsigned td1 = tile_d1;
    unsigned w0 = (2u << 16)                                     // data_size = 4B
                | ((unsigned)pad_en << 20)
                | ((unsigned)pad_interval_code << 22)
                | ((unsigned)pad_amount_code << 25);
    unsigned w1 = (td0 & 0xffffu) << 16;                         // tensor_dim0[15:0]
    unsigned w2 = (td0 >> 16) | ((td1 & 0xffffu) << 16);         // td0 hi | td1 lo
    unsigned w3 = (td1 >> 16) | (tile_d0 << 16);                 // td1 hi | tile_dim0
    unsigned w4 = tile_d1;                                       // tile_dim1 (dim2=0)
    unsigned w5 = (unsigned)(stride & 0xffffffffu);              // dim0_stride lo
    unsigned w6 = (unsigned)((stride >> 32) & 0xffffu);          // dim0_stride hi
    unsigned w7 = 0u;                                            // dim1_stride = 0

    v8i_t g1;
    g1[0] = (int)w0; g1[1] = (int)w1; g1[2] = (int)w2; g1[3] = (int)w3;
    g1[4] = (int)w4; g1[5] = (int)w5; g1[6] = (int)w6; g1[7] = (int)w7;
    v4i_t g2 = {0, 0, 0, 0};
    v4i_t g3 = {0, 0, 0, 0};
    v8i_t g4 = {0, 0, 0, 0, 0, 0, 0, 0};
    __builtin_amdgcn_tensor_load_to_lds(g0, g1, g2, g3, g4, 0);
}

// ---------------------------------------------------------------------------
// fp32 WMMA GEMM with TDM-staged, double-buffered LDS panels.
// C(64xN) = act(A(64xK) @ W(KxN) + b1 + b2)
// One wave per block owns a 16-wide column strip and all four 16-row m-tiles.
// Per k-panel (KT=32 rows):
//   - TDM #1: A panel 64 x pk, row stride lda, LDS rows padded to ARS dwords
//   - TDM #2: B panel pk x 16, row stride N (each weight read exactly once)
//   - issue panel p+1, s_wait_tensorcnt 2  ==> panel p resident (in-order)
//   - inner loop: pure ds_load fragments + v_wmma_f32_16x16x4_f32
// Fragment layouts per CDNA5 ISA 7.12.2.
// ---------------------------------------------------------------------------
template <int ACT>
__global__ void __launch_bounds__(32)
gemm64_wmma(const float* __restrict__ A, int lda,
            const float* __restrict__ W, int N,
            const float* __restrict__ b1, const float* __restrict__ b2,
            float* __restrict__ C, int ldc, int K)
{
    __shared__ float smA[2][64 * ARS];
    __shared__ float smB[2][KT * 16];

    const int lane = threadIdx.x;            // 0..31
    const int l16  = lane & 15;
    const int half = lane >> 4;              // 0 or 1
    const int koff = half * 2;
    const int n0   = blockIdx.x * 16;
    const int col  = n0 + l16;

    v8f acc0 = {}, acc1 = {}, acc2 = {}, acc3 = {};

    const int P = (K + KT - 1) / KT;

    auto issue_panel = [&](int p) {
        const int k0  = p * KT;
        const int pk  = (K - k0 < KT) ? (K - k0) : KT;
        const int buf = p & 1;
        const int pad = (pk == KT) ? 1 : 0;
        // A panel: 64 rows x pk elems, row stride lda; pad rows to ARS dwords
        // (pad codes: interval 4 == 32 dwords, amount 1 == 2 dwords -> 34)
        tdm_load_2d(lds_off(&smA[buf][0]), A + k0, (unsigned)pk, 64u,
                    (unsigned long long)lda, pad, 4, 1);
        // B panel: pk rows x 16 cols, row stride N, densely packed
        tdm_load_2d(lds_off(&smB[buf][0]), W + (size_t)k0 * N + n0, 16u,
                    (unsigned)pk, (unsigned long long)N, 0, 0, 0);
    };

    issue_panel(0);

    for (int p = 0; p < P; ++p) {
        if (p + 1 < P) {
            issue_panel(p + 1);
            __builtin_amdgcn_s_wait_tensorcnt(2);   // panel p complete
        } else {
            __builtin_amdgcn_s_wait_tensorcnt(0);
        }
        asm volatile("" ::: "memory");              // TDM wrote LDS: no load hoisting

        const int buf     = p & 1;
        const int pk      = (K - p * KT < KT) ? (K - p * KT) : KT;
        const int rstride = (pk == KT) ? ARS : pk;  // tail panel is unpadded
        const float* Ab = &smA[buf][0];
        const float* Bb = &smB[buf][0];

        for (int kk = 0; kk < pk; kk += 4) {
            const int ai = kk + koff;
            v2f b;
            b.x = Bb[ai * 16 + l16];
            b.y = Bb[(ai + 1) * 16 + l16];
            v2f a0, a1, a2, a3;
            a0.x = Ab[(l16 +  0) * rstride + ai]; a0.y = Ab[(l16 +  0) * rstride + ai + 1];
            a1.x = Ab[(l16 + 16) * rstride + ai]; a1.y = Ab[(l16 + 16) * rstride + ai + 1];
            a2.x = Ab[(l16 + 32) * rstride + ai]; a2.y = Ab[(l16 + 32) * rstride + ai + 1];
            a3.x = Ab[(l16 + 48) * rstride + ai]; a3.y = Ab[(l16 + 48) * rstride + ai + 1];

            acc0 = __builtin_amdgcn_wmma_f32_16x16x4_f32(false, a0, false, b, (short)0, acc0, false, false);
            acc1 = __builtin_amdgcn_wmma_f32_16x16x4_f32(false, a1, false, b, (short)0, acc1, false, false);
            acc2 = __builtin_amdgcn_wmma_f32_16x16x4_f32(false, a2, false, b, (short)0, acc2, false, false);
            acc3 = __builtin_amdgcn_wmma_f32_16x16x4_f32(false, a3, false, b, (short)0, acc3, false, false);
        }
    }

    float bias = 0.0f;
    if (b1) bias += b1[col];
    if (b2) bias += b2[col];

    const int rbase = half * 8;
#pragma unroll
    for (int i = 0; i < 8; ++i) {
        float v0 = acc0[i] + bias;
        float v1 = acc1[i] + bias;
        float v2 = acc2[i] + bias;
        float v3 = acc3[i] + bias;
        if (ACT == ACT_RELU) {
            v0 = fmaxf(v0, 0.f); v1 = fmaxf(v1, 0.f);
            v2 = fmaxf(v2, 0.f); v3 = fmaxf(v3, 0.f);
        } else if (ACT == ACT_TANH) {
            v0 = tanhf(v0); v1 = tanhf(v1); v2 = tanhf(v2); v3 = tanhf(v3);
        }
        const int r = rbase + i;
        C[(size_t)(r +  0) * ldc + col] = v0;
        C[(size_t)(r + 16) * ldc + col] = v1;
        C[(size_t)(r + 32) * ldc + col] = v2;
        C[(size_t)(r + 48) * ldc + col] = v3;
    }
}

// ---------------------------------------------------------------------------
// Small helper kernels
// ---------------------------------------------------------------------------

// prior filter via lgammaf (beta-binomial pmf), 11 taps
__global__ void prior_kernel(float* __restrict__ prior)
{
    int x = threadIdx.x;
    if (x < PRIOR_K) {
        const float a = 0.1f, b = 0.9f, nn = 10.0f;
        float fx = (float)x;
        float lp = lgammaf(nn + 1.f) + lgammaf(fx + a) + lgammaf(nn - fx + b) + lgammaf(a + b)
                 - (lgammaf(fx + 1.f) + lgammaf(nn - fx + 1.f) + lgammaf(a) + lgammaf(b)
                    + lgammaf(nn + a + b));
        prior[x] = expf(lp);
    }
}

// xin = concat(input_dec(64x80), spkr(64x64)) -> (64x144)
__global__ void build_xin(const float* __restrict__ dec, const float* __restrict__ spkr,
                          float* __restrict__ xin)
{
    int n = blockIdx.x, j = threadIdx.x;  // block 144
    xin[n * 144 + j] = (j < O_DEC) ? dec[n * O_DEC + j] : spkr[n * SPKR + (j - O_DEC)];
}

// GRU gates (hidden state is zero; gh = bhh). out_att_rnn = (1-z)*tanh(gi3 + r*bhh3)
__global__ void gru_gate_kernel(const float* __restrict__ gi, const float* __restrict__ bhh,
                                float* __restrict__ out)
{
    int n = blockIdx.x, j = threadIdx.x;  // block 256
    const float* g = gi + (size_t)n * 3 * H_ATT;
    float r  = sigmf(g[j] + bhh[j]);
    float z  = sigmf(g[H_ATT + j] + bhh[H_ATT + j]);
    float ng = tanhf(g[2 * H_ATT + j] + r * bhh[2 * H_ATT + j]);
    out[n * H_ATT + j] = (1.0f - z) * ng;
}

// Attention logits + softmax over the 10 masked positions (t = 0..min(9,len-1)).
// One block (256 threads = H_ATT) per batch row; deterministic tree reductions.
__global__ void __launch_bounds__(256)
attn_kernel(const float* __restrict__ dcf,       // (64, 336) = (64,16,21)
            const float* __restrict__ dca_proj,  // (16,256)
            const float* __restrict__ pcw,       // (16,31)
            const float* __restrict__ pcb,       // (16,)
            const float* __restrict__ ppw,       // (16,256)
            const float* __restrict__ spkr,      // (64,64)
            const float* __restrict__ spkr_w,    // (64,256)
            const float* __restrict__ speed,     // (64,)
            const float* __restrict__ speed_w,   // (256,)
            const float* __restrict__ apw,       // (256,)
            const float* __restrict__ apb,       // (1,)
            const float* __restrict__ prior,     // (11,)
            const int*   __restrict__ lengths,   // (64,)
            float* __restrict__ att_w)           // (64,16), slots 10..15 zero
{
    const int n = blockIdx.x, h = threadIdx.x;
    __shared__ float red[256];
    __shared__ float slog[10];

    // att_bias_spkr + in_att_speed (per h, shared across t)
    float sb = 0.0f;
    for (int s = 0; s < SPKR; ++s) sb += spkr[n * SPKR + s] * spkr_w[s * H_ATT + h];
    sb = sb / (1.0f + fabsf(sb));                    // soft_sign
    const float basev = sb + speed[n] * speed_w[h];
    const float aw = apw[h];

    for (int t = 0; t < 10; ++t) {
        float e = basev;
        for (int c = 0; c < C_DCA; ++c) {
            e += (pcw[c * PREV_K + (15 - t)] + pcb[c]) * ppw[c * H_ATT + h]; // prev path
            e += dcf[n * (C_DCA * DCA_K) + c * DCA_K + (10 - t)] * dca_proj[c * H_ATT + h];
        }
        red[h] = tanhf(e) * aw;
        __syncthreads();
        for (int s = 128; s > 0; s >>= 1) {
            if (h < s) red[h] += red[h + s];
            __syncthreads();
        }
        if (h == 0) slog[t] = red[0];
        __syncthreads();
    }

    if (h == 0) {
        int len  = lengths[n];
        int tmax = (len < 10) ? len : 10;
        float lg[10], w[10];
        float m = -1e30f;
        for (int t = 0; t < tmax; ++t) {
            float ip = (t <= 5) ? prior[5 - t] : 0.0f;
            ip = fmaxf(ip, 1e-6f);
            lg[t] = slog[t] + apb[0] + logf(ip);
            m = fmaxf(m, lg[t]);
        }
        float sum = 0.0f;
        for (int t = 0; t < tmax; ++t) { w[t] = expf(lg[t] - m); sum += w[t]; }
        sum = fmaxf(sum, 1e-12f);
        for (int t = 0; t < 16; ++t) att_w[n * 16 + t] = (t < tmax) ? w[t] / sum : 0.0f;
    }
}

// spd hidden: relu(speed*w + b), (64x512)
__global__ void spd1_kernel(const float* __restrict__ speed, const float* __restrict__ w,
                            const float* __restrict__ b, float* __restrict__ out)
{
    int n = blockIdx.x, j = threadIdx.x;  // block 512
    out[n * H_DEC + j] = fmaxf(speed[n] * w[j] + b[j], 0.0f);
}

// context[n,h] = sum_t<10 att_w[n,t] * (input_enc[n,t,h] + spd[n,h])
__global__ void context_kernel(const float* __restrict__ att_w, const float* __restrict__ enc,
                               const float* __restrict__ spd, float* __restrict__ ctx)
{
    int n = blockIdx.x, j = threadIdx.x;  // block 512
    float sp = spd[n * H_ENC + j];
    float s = 0.0f;
    for (int t = 0; t < 10; ++t) {
        float a = att_w[n * 16 + t];
        s += a * (enc[((size_t)n * T_ENC + t) * H_ENC + j] + sp);
    }
    ctx[n * H_ENC + j] = s;
}

// in_lstm = concat(out_prenet 512, context 512, spkr 64) -> (64,1088)
__global__ void build_inlstm(const float* __restrict__ pre, const float* __restrict__ ctx,
                             const float* __restrict__ spkr, float* __restrict__ out)
{
    int n = blockIdx.x;
    for (int j = threadIdx.x; j < 1088; j += blockDim.x) {
        float v = (j < 512) ? pre[n * 512 + j]
                : (j < 1024) ? ctx[n * 512 + (j - 512)]
                : spkr[n * SPKR + (j - 1024)];
        out[n * 1088 + j] = v;
    }
}

// LSTM-style gate: out = o * tanh(i * g)  (f gate computed but unused in reference)
__global__ void lstm_gate_kernel(const float* __restrict__ g, float* __restrict__ out)
{
    int n = blockIdx.x;
    int j = blockIdx.y * blockDim.x + threadIdx.x;   // j < 2048
    const float* gr = g + (size_t)n * 4 * LSTM_H;
    float i  = sigmf(gr[j]);
    float gg = tanhf(gr[2 * LSTM_H + j]);
    float o  = sigmf(gr[3 * LSTM_H + j]);
    out[(size_t)n * LSTM_H + j] = o * tanhf(i * gg);
}

// dec_out = concat(out_lstm 2048, in_lstm 1088) -> (64,3136), written into d_out slice
__global__ void build_decout(const float* __restrict__ ol, const float* __restrict__ il,
                             float* __restrict__ out)
{
    int n = blockIdx.x;
    for (int j = threadIdx.x; j < 3136; j += blockDim.x)
        out[(size_t)n * 3136 + j] = (j < 2048) ? ol[(size_t)n * 2048 + j]
                                               : il[n * 1088 + (j - 2048)];
}

// h = LayerNorm(h + sa) * g + b over last dim (512). One block per row.
__global__ void __launch_bounds__(256)
add_ln_kernel(float* __restrict__ h, const float* __restrict__ sa,
              const float* __restrict__ g, const float* __restrict__ b)
{
    int n = blockIdx.x, t = threadIdx.x;
    __shared__ float red[256];
    float x0 = h[n * 512 + t]       + sa[n * 512 + t];
    float x1 = h[n * 512 + 256 + t] + sa[n * 512 + 256 + t];
    red[t] = x0 + x1;
    __syncthreads();
    for (int s = 128; s > 0; s >>= 1) { if (t < s) red[t] += red[t + s]; __syncthreads(); }
    float m = red[0] * (1.0f / 512.0f);
    __syncthreads();
    float d0 = x0 - m, d1 = x1 - m;
    red[t] = d0 * d0 + d1 * d1;
    __syncthreads();
    for (int s = 128; s > 0; s >>= 1) { if (t < s) red[t] += red[t + s]; __syncthreads(); }
    float rstd = rsqrtf(red[0] * (1.0f / 512.0f) + 1e-5f);
    h[n * 512 + t]       = d0 * rstd * g[t]       + b[t];
    h[n * 512 + 256 + t] = d1 * rstd * g[t + 256] + b[t + 256];
}

// ---------------------------------------------------------------------------
// Host side
// ---------------------------------------------------------------------------
static inline void launch_gemm(hipStream_t s, int act,
                               const float* A, int lda, const float* W, int N,
                               const float* b1, const float* b2,
                               float* C, int ldc, int K)
{
    dim3 grid(N / 16), blk(32);
    if (act == ACT_RELU)
        gemm64_wmma<ACT_RELU><<<grid, blk, 0, s>>>(A, lda, W, N, b1, b2, C, ldc, K);
    else if (act == ACT_TANH)
        gemm64_wmma<ACT_TANH><<<grid, blk, 0, s>>>(A, lda, W, N, b1, b2, C, ldc, K);
    else
        gemm64_wmma<ACT_NONE><<<grid, blk, 0, s>>>(A, lda, W, N, b1, b2, C, ldc, K);
}

// input indices (setup_inputs dict order, params flattened in insertion order)
enum {
    IN_INPUT_ENC = 0, IN_INPUT_DEC, IN_SPKR_VEC, IN_LENGTHS, IN_SPEED,
    IN_SPKR_W, IN_PREV_CONV_W, IN_PREV_CONV_B, IN_PREV_PROJ_W,
    IN_ATT_PROJ_W, IN_ATT_PROJ_B, IN_SPEED_W,
    IN_SPD1_W, IN_SPD1_B, IN_SPD2_W, IN_SPD2_B,
    IN_PRE1_W, IN_PRE1_B, IN_PRE2_W, IN_PRE2_B,
    IN_GRU_WIH, IN_GRU_BIH, IN_GRU_BHH,
    IN_DCA1_W, IN_DCA1_B, IN_DCA2_W, IN_DCA_PROJ_W,
    IN_LSTM0_WIH, IN_LSTM0_BIH, IN_LSTM0_BHH,
    IN_LSTM1_WIH, IN_LSTM1_BIH, IN_LSTM1_BHH,
    IN_OL1_W, IN_OL1_B,
    IN_TBASE = 35,          // 12 params per transformer layer
    IN_OL2_W = 59, IN_OL2_B = 60
};
// per-layer param offsets inside a transformer block
enum { TP_QKV_W = 0, TP_QKV_B, TP_OUT_W, TP_OUT_B, TP_LN1_G, TP_LN1_B,
       TP_LN2_G, TP_LN2_B, TP_FF1_W, TP_FF1_B, TP_FF2_W, TP_FF2_B };

extern "C" void kernel_launch(void* const* d_in, const int* in_sizes, int n_in,
                              void* d_out, int out_size, void* d_ws, size_t ws_size,
                              hipStream_t stream)
{
    (void)in_sizes; (void)n_in; (void)out_size; (void)ws_size;
    const float* F[64];
    for (int i = 0; i < 61; ++i) F[i] = (const float*)d_in[i];
    const int* lengths = (const int*)d_in[IN_LENGTHS];

    float* out = (float*)d_out;
    float* out_output  = out;                 // (64, 2, 80)  = 10240
    float* out_context = out + 10240;         // (64, 512)    = 32768
    float* out_decout  = out + 43008;         // (64, 3136)   = 200704

    // workspace layout (floats)
    float* ws = (float*)d_ws;
    size_t o = 0;
    auto alloc = [&](size_t nf) { float* p = ws + o; o += nf; return p; };
    float* xin     = alloc((size_t)NB * 144);
    float* h1      = alloc((size_t)NB * 1024);
    float* prenet  = alloc((size_t)NB * 512);
    float* gi      = alloc((size_t)NB * 768);
    float* attrnn  = alloc((size_t)NB * 256);
    float* dcat1   = alloc((size_t)NB * 256);
    float* dcf     = alloc((size_t)NB * 336);
    float* prior   = alloc(16);
    float* att_w   = alloc((size_t)NB * 16);
    float* spd_h   = alloc((size_t)NB * 512);
    float* spd     = alloc((size_t)NB * 512);
    float* in_lstm = alloc((size_t)NB * 1088);
    float* g0      = alloc((size_t)NB * 8192);
    float* xl1     = alloc((size_t)NB * 2048);
    float* g1      = alloc((size_t)NB * 8192);
    float* outlstm = alloc((size_t)NB * 2048);
    float* hbuf    = alloc((size_t)NB * 512);
    float* qkv     = alloc((size_t)NB * 1536);
    float* sa      = alloc((size_t)NB * 512);
    float* ff1     = alloc((size_t)NB * 1024);
    float* ff2     = alloc((size_t)NB * 512);

    // ---- prior filter (tiny) ----
    prior_kernel<<<1, 16, 0, stream>>>(prior);

    // ---- prenet: x = relu(concat(dec,spkr)@pre1 + b); prenet = relu(x@pre2 + b)
    build_xin<<<NB, 144, 0, stream>>>(F[IN_INPUT_DEC], F[IN_SPKR_VEC], xin);
    launch_gemm(stream, ACT_RELU, xin, 144, F[IN_PRE1_W], 1024, F[IN_PRE1_B], nullptr, h1, 1024, 144);
    launch_gemm(stream, ACT_RELU, h1, 1024, F[IN_PRE2_W], 512, F[IN_PRE2_B], nullptr, prenet, 512, 1024);

    // ---- attention GRU (context0 == 0, so only first 512 rows of gru_wih matter)
    launch_gemm(stream, ACT_NONE, prenet, 512, F[IN_GRU_WIH], 768, F[IN_GRU_BIH], nullptr, gi, 768, 512);
    gru_gate_kernel<<<NB, 256, 0, stream>>>(gi, F[IN_GRU_BHH], attrnn);

    // ---- DCA dynamic filters
    launch_gemm(stream, ACT_TANH, attrnn, 256, F[IN_DCA1_W], 256, F[IN_DCA1_B], nullptr, dcat1, 256, 256);
    launch_gemm(stream, ACT_NONE, dcat1, 256, F[IN_DCA2_W], 336, nullptr, nullptr, dcf, 336, 256);

    // ---- attention weights over the 10 masked positions
    attn_kernel<<<NB, 256, 0, stream>>>(dcf, F[IN_DCA_PROJ_W], F[IN_PREV_CONV_W], F[IN_PREV_CONV_B],
                                        F[IN_PREV_PROJ_W], F[IN_SPKR_VEC], F[IN_SPKR_W],
                                        F[IN_SPEED], F[IN_SPEED_W], F[IN_ATT_PROJ_W],
                                        F[IN_ATT_PROJ_B], prior, lengths, att_w);

    // ---- speed MLP and context (writes the context output slice directly)
    spd1_kernel<<<NB, 512, 0, stream>>>(F[IN_SPEED], F[IN_SPD1_W], F[IN_SPD1_B], spd_h);
    launch_gemm(stream, ACT_TANH, spd_h, 512, F[IN_SPD2_W], 512, F[IN_SPD2_B], nullptr, spd, 512, 512);
    context_kernel<<<NB, 512, 0, stream>>>(att_w, F[IN_INPUT_ENC], spd, out_context);

    // ---- LSTM-style stack (feedforward, no recurrence in this step)
    build_inlstm<<<NB, 256, 0, stream>>>(prenet, out_context, F[IN_SPKR_VEC], in_lstm);
    launch_gemm(stream, ACT_NONE, in_lstm, 1088, F[IN_LSTM0_WIH], 8192,
                F[IN_LSTM0_BIH], F[IN_LSTM0_BHH], g0, 8192, 1088);
    lstm_gate_kernel<<<dim3(NB, 8), 256, 0, stream>>>(g0, xl1);
    launch_gemm(stream, ACT_NONE, xl1, 2048, F[IN_LSTM1_WIH], 8192,
                F[IN_LSTM1_BIH], F[IN_LSTM1_BHH], g1, 8192, 2048);
    lstm_gate_kernel<<<dim3(NB, 8), 256, 0, stream>>>(g1, outlstm);

    // ---- dec_out output slice, then ol1 projection
    build_decout<<<NB, 256, 0, stream>>>(outlstm, in_lstm, out_decout);
    launch_gemm(stream, ACT_NONE, out_decout, 3136, F[IN_OL1_W], 512, F[IN_OL1_B], nullptr,
                hbuf, 512, 3136);

    // ---- 2 transformer-ish layers (self-attn degenerates to V-projection)
    for (int l = 0; l < 2; ++l) {
        const float* const* P = &F[IN_TBASE + l * 12];
        launch_gemm(stream, ACT_NONE, hbuf, 512, P[TP_QKV_W], 1536, P[TP_QKV_B], nullptr,
                    qkv, 1536, 512);
        launch_gemm(stream, ACT_NONE, qkv + 2 * D_T, 1536, P[TP_OUT_W], 512, P[TP_OUT_B], nullptr,
                    sa, 512, 512);
        add_ln_kernel<<<NB, 256, 0, stream>>>(hbuf, sa, P[TP_LN1_G], P[TP_LN1_B]);
        launch_gemm(stream, ACT_RELU, hbuf, 512, P[TP_FF1_W], 1024, P[TP_FF1_B], nullptr,
                    ff1, 1024, 512);
        launch_gemm(stream, ACT_NONE, ff1, 1024, P[TP_FF2_W], 512, P[TP_FF2_B], nullptr,
                    ff2, 512, 1024);
        add_ln_kernel<<<NB, 256, 0, stream>>>(hbuf, ff2, P[TP_LN2_G], P[TP_LN2_B]);
    }

    // ---- final projection straight into the output slice (64 x 160)
    launch_gemm(stream, ACT_NONE, hbuf, 512, F[IN_OL2_W], 160, F[IN_OL2_B], nullptr,
                out_output, 160, 512);
}